// BSplineBasis_45887430590712
// MI455X (gfx1250) — compile-verified
//
#include <hip/hip_runtime.h>
#include <stdint.h>

// B-spline basis fill, CDNA5 (gfx1250).
// T = 262144 rows, 512 cols fp32 out (512 MiB). Each row: <=4 nonzeros at
// columns j-3..j. Bulk zeros streamed via async LDS->global B128 stores
// (ASYNCcnt); value chunks via non-temporal global_store_b128.

#define T_N    262144
#define NKNOTS 516        // K + DEGREE + 1 = 512 + 3 + 1
#define NCOLS  512
#define EPSV   1e-6f
#define ROWS_PER_WAVE 8   // 8 waves/block * 8 rows -> 64 rows per block

typedef float v4f __attribute__((ext_vector_type(4)));

__global__ __launch_bounds__(256)
void BSplineBasis_kernel(const float* __restrict__ tarr,
                         const float* __restrict__ knots,
                         float* __restrict__ out) {
    __shared__ float s_zero[128];     // 512 B of zeros: async-store source
    __shared__ float s_kv[NKNOTS];    // knot vector, LDS-cached

    const int tid  = threadIdx.x;
    const int lane = tid & 31;
    const int wave = tid >> 5;

    for (int i = tid; i < 128;    i += 256) s_zero[i] = 0.0f;
    for (int i = tid; i < NKNOTS; i += 256) s_kv[i]   = knots[i];
    __syncthreads();

    // LDS byte address of this lane's 16B zero chunk (low 32 bits of the
    // flat LDS-aperture address == allocation-relative LDS offset).
    const unsigned lds_zero = (unsigned)(uintptr_t)(&s_zero[lane * 4]);

    const long long row0 =
        ((long long)blockIdx.x * 8 + wave) * ROWS_PER_WAVE;

    for (int rr = 0; rr < ROWS_PER_WAVE; ++rr) {
        const long long row = row0 + rr;
        if (row >= T_N) break;

        const float t = tarr[row];     // exact reference t (broadcast load)

        // j = last index with kv[j] <= t  (kv[0] = -1 <= t always holds).
        // Invariant: kv[lo] <= t < kv[hi] with virtual kv[516] = +inf.
        int lo = 0, hi = NKNOTS;
        while (hi - lo > 1) {
            int mid = (lo + hi) >> 1;
            if (s_kv[mid] <= t) lo = mid; else hi = mid;
        }
        const int  j    = lo;
        const bool zrow = (j > 511);        // t == 1.0 -> all-zero row
        const int  jc   = zrow ? 511 : j;   // clamp for safe knot loads

        float kvL[8];                       // kv[jc-3 .. jc+4]
        #pragma unroll
        for (int q = 0; q < 8; ++q) kvL[q] = s_kv[jc - 3 + q];

        // Local Cox-de Boor band: prev[r] = bases_d[jc - d + r].
        // Mirrors reference's (den >= EPS ? x/den : 0) guards exactly.
        float prev[4] = {1.0f, 0.0f, 0.0f, 0.0f};
        #pragma unroll
        for (int d = 1; d <= 3; ++d) {
            float nxt[4] = {0.0f, 0.0f, 0.0f, 0.0f};
            #pragma unroll
            for (int r = 0; r <= d; ++r) {
                const int   il   = 3 - d + r;                 // kvL index of kv[i]
                const float den1 = kvL[il + d]     - kvL[il];
                const float den2 = kvL[il + d + 1] - kvL[il + 1];
                const float c1 = (den1 >= EPSV) ? (t - kvL[il]) / den1 : 0.0f;
                const float c2 = (den2 >= EPSV) ? (kvL[il + d + 1] - t) / den2 : 0.0f;
                const float a  = (r >= 1)     ? prev[r - 1] : 0.0f;  // bases_{d-1}[i]
                const float b  = (r <= d - 1) ? prev[r]     : 0.0f;  // bases_{d-1}[i+1]
                nxt[r] = c1 * a + c2 * b;
            }
            #pragma unroll
            for (int r = 0; r < 4; ++r) prev[r] = nxt[r];
        }
        // prev[0..3] = basis values at columns j-3 .. j (unused if zrow).

        const int chunkA = zrow ? -1 : ((j - 3) >> 2);   // 16B chunks holding values
        const int chunkB = zrow ? -1 : (j >> 2);

        float* const rowp = out + row * (long long)NCOLS;

        #pragma unroll
        for (int s = 0; s < 4; ++s) {
            const int ci = s * 32 + lane;        // lane-contiguous chunk index
            float* const gp = rowp + ci * 4;
            if (ci == chunkA || ci == chunkB) {
                // Chunk containing (some of) the 4 nonzero columns.
                v4f v;
                #pragma unroll
                for (int q = 0; q < 4; ++q) {
                    const int r = (ci * 4 + q) - (j - 3);
                    float x = 0.0f;
                    x = (r == 0) ? prev[0] : x;
                    x = (r == 1) ? prev[1] : x;
                    x = (r == 2) ? prev[2] : x;
                    x = (r == 3) ? prev[3] : x;
                    v[q] = x;
                }
                __builtin_nontemporal_store(v, (v4f*)gp);  // global_store_b128 NT
            } else {
                // Pure-zero chunk: async engine streams 16B from immutable
                // LDS zeros -> global. No VGPR data, no per-store wait.
                const unsigned long long ga = (unsigned long long)(uintptr_t)gp;
                asm volatile("global_store_async_from_lds_b128 %0, %1, off"
                             :: "v"(ga), "v"(lds_zero) : "memory");
            }
        }
    }

    // Drain async stores (S_ENDPGM also implies wait-idle; explicit for clarity).
    __builtin_amdgcn_s_wait_asynccnt(0);
}

extern "C" void kernel_launch(void* const* d_in, const int* in_sizes, int n_in,
                              void* d_out, int out_size, void* d_ws, size_t ws_size,
                              hipStream_t stream) {
    const float* t_dev  = (const float*)d_in[0];   // 't'     : 262144 fp32
    const float* k_dev  = (const float*)d_in[1];   // 'knots' : 516 fp32
    float*       out    = (float*)d_out;           // 262144 x 512 fp32

    // 64 rows per 256-thread block (8 wave32 x 8 rows) -> 4096 blocks.
    const int blocks = T_N / (8 * ROWS_PER_WAVE);
    BSplineBasis_kernel<<<dim3(blocks), dim3(256), 0, stream>>>(t_dev, k_dev, out);
}